// res_gnn_backbone_6322191859756
// MI455X (gfx1250) — compile-verified
//
#include <hip/hip_runtime.h>

#define DIM 128
#define LSTRIDE 132
#define EPS_LN 1e-5f
#define SLOPE 0.01f

typedef float v2f __attribute__((ext_vector_type(2)));
typedef float v8f __attribute__((ext_vector_type(8)));

// ---------------- init: copy y -> ycur, zero deg + reduction scalars ----------------
__global__ __launch_bounds__(256) void init_kernel(const float* __restrict__ y,
                                                   float* __restrict__ ycur,
                                                   float* __restrict__ deg,
                                                   float* __restrict__ red,
                                                   long long total, int nnodes) {
  long long i = (long long)blockIdx.x * blockDim.x + threadIdx.x;
  if (i < total) ycur[i] = y[i];
  if (i < nnodes) deg[i] = 0.0f;
  if (i < 8) red[i] = 0.0f;
}

// ---------------- in-degree scatter ----------------
__global__ __launch_bounds__(256) void deg_kernel(const long long* __restrict__ dst,
                                                  float* __restrict__ deg, long long E) {
  long long e = (long long)blockIdx.x * blockDim.x + threadIdx.x;
  if (e < E) atomicAdd(&deg[dst[e]], 1.0f);
}

// ---------------- fused triple GEMM: a = yW1+b1 ; h = yW3+b3 - deg*(yW2) ----------------
// One block = 16-row strip of y staged in LDS; 8 waves, each owns a 16-col tile
// and runs three f32 WMMA accumulator chains over K=128 (32 x V_WMMA_F32_16X16X4_F32).
__global__ __launch_bounds__(256) void gemm3_kernel(
    const float* __restrict__ y,
    const float* __restrict__ W1, const float* __restrict__ b1,
    const float* __restrict__ W2,
    const float* __restrict__ W3, const float* __restrict__ b3,
    const float* __restrict__ deg,
    float* __restrict__ outA, float* __restrict__ outH, int nnodes) {
  __shared__ float ytile[16 * LSTRIDE];
  const int m0 = blockIdx.x * 16;
  const int tid = threadIdx.x;

  // cooperative load of 16x128 y strip (zero-fill past end)
  for (int i = tid; i < 16 * DIM; i += 256) {
    int r = i >> 7, c = i & 127;
    int gm = m0 + r;
    ytile[r * LSTRIDE + c] = (gm < nnodes) ? y[(size_t)gm * DIM + c] : 0.0f;
  }
  __syncthreads();

  const int wave  = tid >> 5;
  const int lane  = tid & 31;
  const int mloc  = lane & 15;          // A: row
  const int khalf = (lane >> 4) << 1;   // A/B: K sub-offset (0 or 2)
  const int nloc  = lane & 15;          // B/C/D: col
  const int nc    = wave * 16 + nloc;   // global output column (0..127)

  v8f accA = {}, accB = {}, accC = {};
  for (int k0 = 0; k0 < DIM; k0 += 4) {
    const int ka = k0 + khalf;
    v2f af;
    af.x = ytile[mloc * LSTRIDE + ka];
    af.y = ytile[mloc * LSTRIDE + ka + 1];
    const float* w1p = W1 + (size_t)ka * DIM + nc;
    const float* w2p = W2 + (size_t)ka * DIM + nc;
    const float* w3p = W3 + (size_t)ka * DIM + nc;
    v2f bf1, bf2, bf3;
    bf1.x = w1p[0]; bf1.y = w1p[DIM];
    bf2.x = w2p[0]; bf2.y = w2p[DIM];
    bf3.x = w3p[0]; bf3.y = w3p[DIM];
    accA = __builtin_amdgcn_wmma_f32_16x16x4_f32(false, af, false, bf1, (short)0, accA, false, false);
    accB = __builtin_amdgcn_wmma_f32_16x16x4_f32(false, af, false, bf2, (short)0, accB, false, false);
    accC = __builtin_amdgcn_wmma_f32_16x16x4_f32(false, af, false, bf3, (short)0, accC, false, false);
  }

  // D layout: v[i] holds row (8*(lane/16) + i), col = lane%16
  const int mbase = (lane >> 4) << 3;
  const float bias1 = b1[nc];
  const float bias3 = b3[nc];
#pragma unroll
  for (int i = 0; i < 8; ++i) {
    int m = m0 + mbase + i;
    if (m < nnodes) {
      float dg = deg[m];
      outA[(size_t)m * DIM + nc] = accA[i] + bias1;
      outH[(size_t)m * DIM + nc] = accC[i] + bias3 - dg * accB[i];
    }
  }
}

// ---------------- edge scatter: h[dst] += a[src]; one wave per edge, float4 per lane ----------------
__global__ __launch_bounds__(256) void scatter_kernel(const float* __restrict__ a,
                                                      const long long* __restrict__ src,
                                                      const long long* __restrict__ dst,
                                                      float* __restrict__ h, long long E) {
  long long e = (long long)blockIdx.x * 8 + (threadIdx.x >> 5);
  if (e >= E) return;
  int c = (threadIdx.x & 31) << 2;
  long long s = src[e], d = dst[e];
  const float4 v = *reinterpret_cast<const float4*>(a + s * DIM + c);
  float* hp = h + d * DIM + c;
  atomicAdd(hp + 0, v.x);
  atomicAdd(hp + 1, v.y);
  atomicAdd(hp + 2, v.z);
  atomicAdd(hp + 3, v.w);
}

// ---------------- grid-wide sum / sumsq reduction into red[0], red[1] ----------------
__global__ __launch_bounds__(256) void reduce_kernel(const float* __restrict__ h,
                                                     float* __restrict__ red, long long total) {
  float s = 0.f, s2 = 0.f;
  const long long stride = (long long)gridDim.x * blockDim.x;
  for (long long i = (long long)blockIdx.x * blockDim.x + threadIdx.x; i < total; i += stride) {
    float v = h[i];
    s += v;
    s2 += v * v;
  }
#pragma unroll
  for (int off = 16; off > 0; off >>= 1) {
    s  += __shfl_xor(s, off, 32);
    s2 += __shfl_xor(s2, off, 32);
  }
  __shared__ float ls[8], ls2[8];
  int wave = threadIdx.x >> 5, lane = threadIdx.x & 31;
  if (lane == 0) { ls[wave] = s; ls2[wave] = s2; }
  __syncthreads();
  if (threadIdx.x == 0) {
    float ts = 0.f, ts2 = 0.f;
#pragma unroll
    for (int w = 0; w < 8; ++w) { ts += ls[w]; ts2 += ls2[w]; }
    atomicAdd(&red[0], ts);
    atomicAdd(&red[1], ts2);
  }
}

// ---------------- normalize + affine + leaky_relu + residual ----------------
__global__ __launch_bounds__(256) void finalize_kernel(const float* __restrict__ h,
                                                       const float* __restrict__ red,
                                                       const float* __restrict__ yin,
                                                       const float* __restrict__ lnw,
                                                       const float* __restrict__ lnb,
                                                       float* __restrict__ yout,
                                                       long long total) {
  long long i = (long long)blockIdx.x * blockDim.x + threadIdx.x;
  if (i >= total) return;
  const float invn = 1.0f / (float)total;
  const float mean = red[0] * invn;
  float var = red[1] * invn - mean * mean;
  var = var > 0.f ? var : 0.f;
  const float inv = 1.0f / (sqrtf(var) + EPS_LN);
  const int c = (int)(i & 127);
  float v = (h[i] - mean) * inv * lnw[c] + lnb[c];
  v = (v >= 0.f) ? v : SLOPE * v;
  yout[i] = yin[i] + v;
}

extern "C" void kernel_launch(void* const* d_in, const int* in_sizes, int n_in,
                              void* d_out, int out_size, void* d_ws, size_t ws_size,
                              hipStream_t stream) {
  const float*     y   = (const float*)d_in[0];
  const long long* ei  = (const long long*)d_in[1];  // int64 (2, E): src rows then dst rows
  const float*     W1  = (const float*)d_in[2];
  const float*     b1  = (const float*)d_in[3];
  const float*     W2  = (const float*)d_in[4];
  const float*     W3  = (const float*)d_in[5];
  const float*     b3  = (const float*)d_in[6];
  const float*     lnw = (const float*)d_in[7];
  const float*     lnb = (const float*)d_in[8];

  const int nnodes     = in_sizes[0] / DIM;
  const long long E    = (long long)in_sizes[1] / 2;
  const long long total = (long long)nnodes * DIM;
  const long long* src = ei;
  const long long* dst = ei + E;

  // workspace layout (floats): deg[N] | red[8] | a[N*D] | h[N*D] | ycur[N*D]
  float* ws   = (float*)d_ws;
  float* deg  = ws;
  float* red  = ws + nnodes;
  float* bufA = ws + nnodes + 8;
  float* bufH = bufA + total;
  float* ycur = bufH + total;

  dim3 blk(256);
  const int gElem = (int)((total + 255) / 256);
  const int gDeg  = (int)((E + 255) / 256);
  const int gGemm = (nnodes + 15) / 16;
  const int gScat = (int)((E + 7) / 8);

  init_kernel<<<gElem, blk, 0, stream>>>(y, ycur, deg, red, total, nnodes);
  deg_kernel<<<gDeg, blk, 0, stream>>>(dst, deg, E);

  for (int l = 0; l < 3; ++l) {
    const float* W1l = W1 + (size_t)l * DIM * DIM;
    const float* W2l = W2 + (size_t)l * DIM * DIM;
    const float* W3l = W3 + (size_t)l * DIM * DIM;
    gemm3_kernel<<<gGemm, blk, 0, stream>>>(ycur, W1l, b1 + l * DIM, W2l, W3l, b3 + l * DIM,
                                            deg, bufA, bufH, nnodes);
    scatter_kernel<<<gScat, blk, 0, stream>>>(bufA, src, dst, bufH, E);
    reduce_kernel<<<1024, blk, 0, stream>>>(bufH, red + 2 * l, total);
    float* yout = (l == 2) ? (float*)d_out : ycur;
    finalize_kernel<<<gElem, blk, 0, stream>>>(bufH, red + 2 * l, ycur,
                                               lnw + l * DIM, lnb + l * DIM, yout, total);
  }
}